// GCN_22351009808409
// MI455X (gfx1250) — compile-verified
//
#include <hip/hip_runtime.h>
#include <hip/hip_bf16.h>
#include <stdint.h>

typedef __attribute__((ext_vector_type(2))) float v2f;
typedef __attribute__((ext_vector_type(8))) float v8f;

#define FEAT 128
#define LDS_PITCH 132   // pad so B-fragment lanes hit distinct LDS bank pairs

// ---------------------------------------------------------------------------
// 1) deg[i] = 1 (self loop)
__global__ void k_init_deg(float* __restrict__ deg, int n) {
    int i = blockIdx.x * blockDim.x + threadIdx.x;
    if (i < n) deg[i] = 1.0f;
}

// 2) deg[dst] += 1 over real edges
__global__ void k_count_deg(const long long* __restrict__ ei, float* __restrict__ deg,
                            long long E) {
    long long e = (long long)blockIdx.x * blockDim.x + threadIdx.x;
    if (e < E) {
        int dst = (int)ei[E + e];
        unsafeAtomicAdd(&deg[dst], 1.0f);
    }
}

// 3) dinv[i] = rsqrt(deg[i])  (deg >= 1 always due to self loop)
__global__ void k_dinv(const float* __restrict__ deg, float* __restrict__ dinv, int n) {
    int i = blockIdx.x * blockDim.x + threadIdx.x;
    if (i < n) dinv[i] = __frsqrt_rn(deg[i]);
}

// ---------------------------------------------------------------------------
// 4) h = x @ W  via V_WMMA_F32_16X16X4_F32 (exact fp32).
//    Block = 256 threads = 8 waves; each wave computes one 16-row strip
//    (16 x 128 of h) as 8 column tiles, K-loop of 32 x k=4 WMMA steps.
__global__ void __launch_bounds__(256)
k_gemm_wmma(const float* __restrict__ x, const float* __restrict__ W,
            float* __restrict__ h, int M) {
    __shared__ float ldsW[FEAT * LDS_PITCH];  // W transposed: ldsW[n*PITCH + k] = W[k][n]

    const int tid = threadIdx.x;
    // cooperative load + transpose of W (128x128 row-major [K][N])
    for (int i = tid; i < FEAT * FEAT; i += 256) {
        int k = i >> 7;
        int n = i & (FEAT - 1);
        ldsW[n * LDS_PITCH + k] = W[i];
    }
    __syncthreads();

    const int wave = tid >> 5;
    const int lane = tid & 31;
    const int row0 = (blockIdx.x * 8 + wave) * 16;
    if (row0 >= M) return;               // wave-uniform: EXEC all-ones below
    const int half = lane >> 4;          // 0: K even pair, 1: K odd pair (per WMMA layout)
    const int l16  = lane & 15;

    v8f acc[8];
#pragma unroll
    for (int t = 0; t < 8; ++t) acc[t] = (v8f){0,0,0,0,0,0,0,0};

    const float* xrow = x + (size_t)(row0 + l16) * FEAT;

    for (int ks = 0; ks < FEAT / 4; ++ks) {
        const int kbase = ks * 4 + half * 2;
        // A fragment: lane (l16, half) holds x[row0+l16][kbase], x[..][kbase+1]
        v2f a;
        a.x = xrow[kbase + 0];
        a.y = xrow[kbase + 1];
#pragma unroll
        for (int t = 0; t < 8; ++t) {
            const int n = t * 16 + l16;
            // B fragment: lane (l16, half) holds W[kbase][n], W[kbase+1][n]
            v2f b;
            b.x = ldsW[n * LDS_PITCH + kbase + 0];
            b.y = ldsW[n * LDS_PITCH + kbase + 1];
            acc[t] = __builtin_amdgcn_wmma_f32_16x16x4_f32(
                false, a, false, b, (short)0, acc[t], false, false);
        }
    }

    // C/D layout: VGPR v -> M = v + 8*half, N = t*16 + l16
#pragma unroll
    for (int t = 0; t < 8; ++t) {
        const int col = t * 16 + l16;
#pragma unroll
        for (int v = 0; v < 8; ++v) {
            h[(size_t)(row0 + v + 8 * half) * FEAT + col] = acc[t][v];
        }
    }
}

// ---------------------------------------------------------------------------
// 5) out[i][:] = h[i][:] * dinv[i]^2   (self-loop term; also initializes d_out)
__global__ void k_init_out(const float* __restrict__ h, const float* __restrict__ dinv,
                           float* __restrict__ out, long long n4) {
    long long i4 = (long long)blockIdx.x * blockDim.x + threadIdx.x;
    if (i4 >= n4) return;
    int row = (int)(i4 >> 5);            // 32 float4 per 128-wide row
    float s = dinv[row];
    s = s * s;
    float4 v = ((const float4*)h)[i4];
    v.x *= s; v.y *= s; v.z *= s; v.w *= s;
    ((float4*)out)[i4] = v;
}

// 6) one wave per edge: out[dst] += h[src] * dinv[src]*dinv[dst]
__global__ void __launch_bounds__(256)
k_scatter(const long long* __restrict__ ei, const float* __restrict__ h,
          const float* __restrict__ dinv, float* __restrict__ out, long long E) {
    const int lane = threadIdx.x & 31;
    const long long e = (long long)blockIdx.x * (blockDim.x >> 5) + (threadIdx.x >> 5);
    if (e >= E) return;
    const int src = (int)ei[e];
    const int dst = (int)ei[E + e];
    const float norm = dinv[src] * dinv[dst];
    float4 v = ((const float4*)(h + (size_t)src * FEAT))[lane];
    float* od = out + (size_t)dst * FEAT + lane * 4;
    unsafeAtomicAdd(od + 0, v.x * norm);
    unsafeAtomicAdd(od + 1, v.y * norm);
    unsafeAtomicAdd(od + 2, v.z * norm);
    unsafeAtomicAdd(od + 3, v.w * norm);
}

// ---------------------------------------------------------------------------
extern "C" void kernel_launch(void* const* d_in, const int* in_sizes, int n_in,
                              void* d_out, int out_size, void* d_ws, size_t ws_size,
                              hipStream_t stream) {
    const float*      x  = (const float*)d_in[0];
    const long long*  ei = (const long long*)d_in[1];
    const float*      W  = (const float*)d_in[2];
    float*            out = (float*)d_out;

    const int       N = in_sizes[0] / FEAT;       // 100000
    const long long E = (long long)in_sizes[1] / 2; // 1600000

    // workspace layout: deg[N] | dinv[N] | h[N*128]  (h offset 800000 B, 16B aligned)
    float* deg  = (float*)d_ws;
    float* dinv = deg + N;
    float* h    = dinv + N;

    const int T = 256;

    k_init_deg<<<(N + T - 1) / T, T, 0, stream>>>(deg, N);
    k_count_deg<<<(unsigned)((E + T - 1) / T), T, 0, stream>>>(ei, deg, E);
    k_dinv<<<(N + T - 1) / T, T, 0, stream>>>(deg, dinv, N);

    // 8 row-strips of 16 per block -> 128 rows per block
    const int strips = (N + 15) / 16;
    const int gemm_blocks = (strips + 7) / 8;
    k_gemm_wmma<<<gemm_blocks, T, 0, stream>>>(x, W, h, N);

    const long long n4 = (long long)N * (FEAT / 4);
    k_init_out<<<(unsigned)((n4 + T - 1) / T), T, 0, stream>>>(h, dinv, out, n4);

    const long long waves_per_block = T / 32;
    k_scatter<<<(unsigned)((E + waves_per_block - 1) / waves_per_block), T, 0, stream>>>(
        ei, h, dinv, out, E);
}